// myMaxPool2d_7988639171216
// MI455X (gfx1250) — compile-verified
//
#include <hip/hip_runtime.h>
#include <hip/hip_bf16.h>

// ---------------------------------------------------------------------------
// MaxPool 2x2 stride 1 VALID on fp32 NCHW 16x96x224x224 -> 16x96x223x223.
// Bandwidth-bound: ~614 MB total DRAM traffic -> ~26us floor @ 23.3 TB/s.
// Strategy: async-stage 17 input rows per block into LDS via
// global_load_async_to_lds_b128 (1.0625x read amp), compute 16 output rows
// from LDS with b128 ds loads, stream results with non-temporal stores.
// ---------------------------------------------------------------------------

#define NPLANES   (16 * 96)   // N*C
#define H         224
#define W         224
#define OH        223
#define OW        223
#define ROWS_OUT  16
#define ROWS_IN   (ROWS_OUT + 1)
#define STRIPS    ((OH + ROWS_OUT - 1) / ROWS_OUT)   // 14
#define SEG_PER_ROW (W / 4)                          // 56 float4 segments/row

#if defined(__has_builtin)
#  if __has_builtin(__builtin_amdgcn_global_load_async_to_lds_b128)
#    define USE_ASYNC_LDS 1
#  else
#    define USE_ASYNC_LDS 0
#  endif
#else
#  define USE_ASYNC_LDS 0
#endif

// Exact pointee type the builtin expects (per hipcc diagnostic):
//   __attribute__((__vector_size__(4 * sizeof(int)))) int  __device__ *
typedef int v4i __attribute__((vector_size(4 * sizeof(int))));
typedef __attribute__((address_space(1))) v4i* gv4i_ptr;   // global (AS1)
typedef __attribute__((address_space(3))) v4i* lv4i_ptr;   // LDS    (AS3)

__global__ __launch_bounds__(256) void maxpool2x2s1_kernel(
    const float* __restrict__ x, float* __restrict__ y) {
  __shared__ __align__(16) float lds[ROWS_IN * W];   // 15232 bytes

  const int blk   = blockIdx.x;
  const int strip = blk % STRIPS;
  const int plane = blk / STRIPS;
  const int r0    = strip * ROWS_OUT;
  const int rows_out = min(ROWS_OUT, OH - r0);       // 16, or 15 (last strip)
  const int rows_in  = rows_out + 1;                 // r0+rows_in <= 224 always

  const float* __restrict__ xin = x + ((size_t)plane * H + (size_t)r0) * W;
  const int t = threadIdx.x;

  // ---- Stage rows_in input rows into LDS (async global->LDS, b128) ----
  const int nseg = rows_in * SEG_PER_ROW;            // <= 952
  for (int seg = t; seg < nseg; seg += 256) {
    const int row  = seg / SEG_PER_ROW;
    const int c4   = (seg % SEG_PER_ROW) * 4;
    const float* gptr = xin + row * W + c4;          // 16B aligned
    float*       lptr = &lds[row * W + c4];          // 16B aligned
#if USE_ASYNC_LDS
    __builtin_amdgcn_global_load_async_to_lds_b128(
        (gv4i_ptr)const_cast<float*>(gptr),
        (lv4i_ptr)lptr,
        /*offset=*/0, /*cpol=*/0);
#else
    *(float4*)lptr = *(const float4*)gptr;
#endif
  }

#if USE_ASYNC_LDS
#  if defined(__has_builtin) && __has_builtin(__builtin_amdgcn_s_wait_asynccnt)
  __builtin_amdgcn_s_wait_asynccnt(0);
#  else
  asm volatile("s_wait_asynccnt 0" ::: "memory");
#  endif
#endif
  __syncthreads();

  // ---- Compute: thread t -> col group lt (4 cols), row slot lr (4 passes) --
  if (t < 4 * SEG_PER_ROW) {                         // 224 active threads
    const int lt = t % SEG_PER_ROW;                  // 0..55
    const int lr = t / SEG_PER_ROW;                  // 0..3
    const int c0 = lt * 4;
    float* __restrict__ yout = y + ((size_t)plane * OH + (size_t)r0) * OW;

    #pragma unroll
    for (int p = 0; p < ROWS_OUT / 4; ++p) {
      const int rr = p * 4 + lr;                     // row within strip
      if (rr < rows_out) {
        const float4 a = *(const float4*)&lds[rr * W + c0];        // row r
        const float4 b = *(const float4*)&lds[(rr + 1) * W + c0];  // row r+1
        float a4 = -__builtin_inff(), b4 = -__builtin_inff();
        if (c0 + 4 < W) {                            // overlap column
          a4 = lds[rr * W + c0 + 4];
          b4 = lds[(rr + 1) * W + c0 + 4];
        }
        const float rm0 = fmaxf(a.x, b.x);
        const float rm1 = fmaxf(a.y, b.y);
        const float rm2 = fmaxf(a.z, b.z);
        const float rm3 = fmaxf(a.w, b.w);
        const float rm4 = fmaxf(a4, b4);
        const float o0 = fmaxf(rm0, rm1);
        const float o1 = fmaxf(rm1, rm2);
        const float o2 = fmaxf(rm2, rm3);
        const float o3 = fmaxf(rm3, rm4);

        float* yr = yout + (size_t)rr * OW + c0;
        const int ncols = min(4, OW - c0);           // 4, or 3 for lt==55
        __builtin_nontemporal_store(o0, yr + 0);
        if (ncols > 1) __builtin_nontemporal_store(o1, yr + 1);
        if (ncols > 2) __builtin_nontemporal_store(o2, yr + 2);
        if (ncols > 3) __builtin_nontemporal_store(o3, yr + 3);
      }
    }
  }
}

extern "C" void kernel_launch(void* const* d_in, const int* in_sizes, int n_in,
                              void* d_out, int out_size, void* d_ws, size_t ws_size,
                              hipStream_t stream) {
  (void)in_sizes; (void)n_in; (void)out_size; (void)d_ws; (void)ws_size;
  const float* x = (const float*)d_in[0];
  float*       y = (float*)d_out;
  const int nblocks = NPLANES * STRIPS;              // 1536 * 14 = 21504
  maxpool2x2s1_kernel<<<nblocks, 256, 0, stream>>>(x, y);
}